// CausalTransition_68066641707234
// MI455X (gfx1250) — compile-verified
//
#include <hip/hip_runtime.h>
#include <hip/hip_bf16.h>

typedef float v2f __attribute__((ext_vector_type(2)));
typedef float v8f __attribute__((ext_vector_type(8)));

#define NEGBIG -1000000000.0f

// ---------------------------------------------------------------------------
// K1: nodes[b,n,c] = pl (latent^T + positional encoding) for n<196, a_b for n=196
// ---------------------------------------------------------------------------
__global__ void init_nodes_kernel(const float* __restrict__ latent,
                                  const float* __restrict__ a_b,
                                  float* __restrict__ nodes) {
  int idx = blockIdx.x * blockDim.x + threadIdx.x;
  const int TOT = 4 * 197 * 64;
  if (idx >= TOT) return;
  int c = idx & 63;
  int n = (idx >> 6) % 197;
  int b = idx / (197 * 64);
  float v;
  if (n == 196) {
    v = a_b[c];
  } else {
    float lat = latent[(b * 64 + c) * 196 + n];
    // div = exp(-(c - c%2) * ln(10000)/64)
    float d = __expf(-(float)(c & ~1) * (9.210340371976184f / 64.0f));
    float ang = (float)n * d;
    float pe = (c & 1) ? __cosf(ang) : __sinf(ang);
    v = lat + pe;
  }
  nodes[idx] = v;
}

// ---------------------------------------------------------------------------
// Generic fp32 WMMA GEMM: per batch b (grid.z): Out = A(MxK) @ W(KxNc) + bias.
// One wave computes a 16x32 slab = two 16x16 WMMA tiles sharing one A frag.
// Out-of-range rows/cols are loaded from clamped (valid) addresses; since
// output (m,n) depends only on A row m and W col n, garbage never leaks into
// valid outputs — bounds are enforced only at store time. Inner loop is
// branch-free: loads + v_wmma_f32_16x16x4_f32 only.
// transOut==1 stores Out[b] as (Nc x M) (transposed), ld = M.  K % 4 == 0.
// ---------------------------------------------------------------------------
__global__ __launch_bounds__(32)
void wmma_gemm_kernel(const float* __restrict__ A,
                      const float* __restrict__ W,
                      const float* __restrict__ bias,
                      float* __restrict__ Out,
                      int M, int K, int Nc,
                      int strideA, int strideOut,
                      int transOut) {
  const int lane = threadIdx.x;           // 0..31
  const int half = lane >> 4;             // 0 or 1
  const int l    = lane & 15;
  const int n0 = blockIdx.x * 32;         // two 16-wide tiles
  const int i0 = blockIdx.y * 16;
  const int b  = blockIdx.z;

  const float* Ab = A + (size_t)b * strideA;
  float* Ob = Out + (size_t)b * strideOut;

  const int arow  = i0 + l;
  const int bcol0 = n0 + l;
  const int bcol1 = n0 + 16 + l;
  // clamped (always-valid) load addresses; garbage stays in dead rows/cols
  const int arow_s  = (arow  < M)  ? arow  : 0;
  const int bcol0_s = (bcol0 < Nc) ? bcol0 : 0;
  const int bcol1_s = (bcol1 < Nc) ? bcol1 : 0;

  const float* aptr  = Ab + (size_t)arow_s * K + 2 * half;   // += 4 per step
  const float* wptr0 = W + (size_t)(2 * half) * Nc + bcol0_s; // += 4*Nc
  const float* wptr1 = W + (size_t)(2 * half) * Nc + bcol1_s;
  const size_t wstep = (size_t)4 * Nc;

  v8f acc0, acc1;
  #pragma unroll
  for (int r = 0; r < 8; ++r) { acc0[r] = 0.0f; acc1[r] = 0.0f; }

  for (int k0 = 0; k0 < K; k0 += 4) {
    v2f af;
    af.x = aptr[0];
    af.y = aptr[1];
    v2f bf0, bf1;
    bf0.x = wptr0[0];
    bf0.y = wptr0[Nc];
    bf1.x = wptr1[0];
    bf1.y = wptr1[Nc];
    acc0 = __builtin_amdgcn_wmma_f32_16x16x4_f32(
        false, af, false, bf0, (short)0, acc0, false, false);
    acc1 = __builtin_amdgcn_wmma_f32_16x16x4_f32(
        false, af, false, bf1, (short)0, acc1, false, false);
    aptr  += 4;
    wptr0 += wstep;
    wptr1 += wstep;
  }

  #pragma unroll
  for (int r = 0; r < 8; ++r) {
    const int m = r + 8 * half;
    const int row = i0 + m;
    if (row < M) {
      const int c0 = n0 + l;
      const int c1 = n0 + 16 + l;
      if (c0 < Nc) {
        float v = acc0[r] + (bias ? bias[c0] : 0.0f);
        if (transOut) Ob[(size_t)c0 * M + row] = v;
        else          Ob[(size_t)row * Nc + c0] = v;
      }
      if (c1 < Nc) {
        float v = acc1[r] + (bias ? bias[c1] : 0.0f);
        if (transOut) Ob[(size_t)c1 * M + row] = v;
        else          Ob[(size_t)row * Nc + c1] = v;
      }
    }
  }
}

// ---------------------------------------------------------------------------
// K3: pairwise edge score + gumbel-hard gate, writes padded adjacency adj_p.
// Block = one (b,i) row. i==196 -> zero pad row. Thread j==196 -> pad col 1.0.
// adj = sigmoid( sum_k leaky(U[b,i,k] + VT[b,k,j]) * w2[k] + b2 )  (b1 folded
// into U by the GEMM bias).
// ---------------------------------------------------------------------------
__global__ __launch_bounds__(256)
void pairwise_kernel(const float* __restrict__ U,      // (B,196,800) incl. b1
                     const float* __restrict__ VT,     // (B,800,196)
                     const float* __restrict__ w2,     // (800)
                     const float* __restrict__ b2p,    // (1)
                     const float* __restrict__ gumbel, // (B,196,196,2)
                     float* __restrict__ adj_p) {      // (B,197,197)
  const int b = blockIdx.x / 197;
  const int i = blockIdx.x % 197;
  const int tid = threadIdx.x;

  if (i == 196) {  // padded last row = 0
    for (int j = tid; j < 197; j += 256)
      adj_p[((size_t)b * 197 + 196) * 197 + j] = 0.0f;
    return;
  }

  __shared__ float Urow[800];
  __shared__ float w2s[800];
  for (int k = tid; k < 800; k += 256) {
    Urow[k] = U[((size_t)b * 196 + i) * 800 + k];
    w2s[k]  = w2[k];
  }
  __syncthreads();

  const int j = tid;
  if (j < 197) {
    float outv;
    if (j == 196) {
      outv = 1.0f;  // padded last column = 1
    } else {
      const float* vt = VT + (size_t)b * 800 * 196 + j;
      float acc = 0.0f;
      #pragma unroll 4
      for (int k = 0; k < 800; ++k) {
        float v = Urow[k] + vt[(size_t)k * 196];
        v = (v >= 0.0f) ? v : 0.01f * v;
        acc = fmaf(v, w2s[k], acc);
      }
      float adj = 1.0f / (1.0f + __expf(-(acc + b2p[0])));
      float l1 = __logf(fmaxf(adj, 1e-4f));
      float l0 = __logf(fmaxf(1.0f - adj, 1e-4f));
      const float* g = gumbel + (((size_t)(b * 196 + i) * 196) + j) * 2;
      float cg = (l1 + g[1] > l0 + g[0]) ? 1.0f : 0.0f;  // hard gumbel argmax
      outv = adj * cg;
    }
    adj_p[((size_t)b * 197 + i) * 197 + j] = outv;
  }
}

// ---------------------------------------------------------------------------
// K5: GATv2 attention. Block = one (b,t). adjT[b,t,s] = adj_p[b,s,t].
// logit[s,h] = sum_o att[h,o]*leaky(xl[b,s,h,o]+xr[b,t,h,o]+a*we[h,o], 0.2)
// alpha = masked softmax over s; out[b,t,:] = leaky_s(sum_s alpha*xl + bias)
// ---------------------------------------------------------------------------
__global__ __launch_bounds__(256)
void gat_attn_kernel(const float* __restrict__ xl,    // (B,Np,H*O)
                     const float* __restrict__ xr,    // (B,Np,H*O)
                     const float* __restrict__ adj_p, // (B,Np,Np)
                     const float* __restrict__ att,   // (H,O)
                     const float* __restrict__ we,    // (H*O)
                     const float* __restrict__ bias,  // (H*O)
                     float* __restrict__ out,         // (B,Np,H*O)
                     int Np, int H, int O, float outSlope) {
  const int b = blockIdx.x / Np;
  const int t = blockIdx.x % Np;
  const int tid = threadIdx.x;
  const int HO = H * O;

  __shared__ float xrt[512];
  __shared__ float wes[512];
  __shared__ float atts[512];
  __shared__ float lg[256];    // logits -> alpha
  __shared__ float red[256];   // reduction scratch

  for (int idx = tid; idx < HO; idx += 256) {
    xrt[idx]  = xr[((size_t)b * Np + t) * HO + idx];
    wes[idx]  = we[idx];
    atts[idx] = att[idx];
  }
  __syncthreads();

  const int s = tid;  // Np <= 256: one s per thread
  float a = 0.0f;
  if (s < Np) a = adj_p[((size_t)b * Np + s) * Np + t];

  for (int h = 0; h < H; ++h) {
    // --- logits ---
    float lgv = NEGBIG;
    if (s < Np && a != 0.0f) {
      const float* xls = xl + ((size_t)b * Np + s) * HO + h * O;
      float sum = 0.0f;
      for (int o = 0; o < O; ++o) {
        float v = xls[o] + xrt[h * O + o] + a * wes[h * O + o];
        v = (v >= 0.0f) ? v : 0.2f * v;
        sum = fmaf(v, atts[h * O + o], sum);
      }
      lgv = sum;
    }
    lg[tid] = lgv;
    red[tid] = lgv;
    __syncthreads();
    // --- max reduce ---
    for (int st = 128; st > 0; st >>= 1) {
      if (tid < st) red[tid] = fmaxf(red[tid], red[tid + st]);
      __syncthreads();
    }
    float mx = red[0];
    __syncthreads();
    // --- exp + sum ---
    float ev = 0.0f;
    if (s < Np && a != 0.0f) ev = __expf(lg[tid] - mx);
    red[tid] = ev;
    __syncthreads();
    for (int st = 128; st > 0; st >>= 1) {
      if (tid < st) red[tid] += red[tid + st];
      __syncthreads();
    }
    float sm = red[0];
    __syncthreads();
    lg[tid] = (sm > 0.0f) ? ev / sm : 0.0f;  // alpha
    __syncthreads();
    // --- weighted aggregation over s ---
    for (int o = tid; o < O; o += 256) {
      float acc = 0.0f;
      const float* xlc = xl + (size_t)b * Np * HO + h * O + o;
      for (int ss = 0; ss < Np; ++ss)
        acc = fmaf(lg[ss], xlc[(size_t)ss * HO], acc);
      float r = acc + bias[h * O + o];
      out[((size_t)b * Np + t) * HO + h * O + o] =
          (r >= 0.0f) ? r : outSlope * r;
    }
    __syncthreads();
  }
}

// ---------------------------------------------------------------------------
// K7: final softmax over C=64 of x2[:, :196, :64]
// ---------------------------------------------------------------------------
__global__ __launch_bounds__(64)
void final_softmax_kernel(const float* __restrict__ x2, float* __restrict__ out) {
  const int b = blockIdx.x / 196;
  const int n = blockIdx.x % 196;
  const int tid = threadIdx.x;  // 0..63
  __shared__ float red[64];
  float v = x2[((size_t)b * 197 + n) * 320 + tid];
  red[tid] = v;
  __syncthreads();
  for (int st = 32; st > 0; st >>= 1) {
    if (tid < st) red[tid] = fmaxf(red[tid], red[tid + st]);
    __syncthreads();
  }
  float mx = red[0];
  __syncthreads();
  float e = __expf(v - mx);
  red[tid] = e;
  __syncthreads();
  for (int st = 32; st > 0; st >>= 1) {
    if (tid < st) red[tid] += red[tid + st];
    __syncthreads();
  }
  float sm = red[0];
  out[((size_t)b * 196 + n) * 64 + tid] = e / sm;
}

// ---------------------------------------------------------------------------
extern "C" void kernel_launch(void* const* d_in, const int* in_sizes, int n_in,
                              void* d_out, int out_size, void* d_ws, size_t ws_size,
                              hipStream_t stream) {
  const float* latent = (const float*)d_in[0];
  const float* gumbel = (const float*)d_in[1];
  const float* d0_w1  = (const float*)d_in[2];
  const float* d0_b1  = (const float*)d_in[3];
  const float* d0_w2  = (const float*)d_in[4];
  const float* d0_b2  = (const float*)d_in[5];
  // d_in[6..9]: d1_* MLP — multiplied by mask==0 in the reference, unused.
  // d_in[10]: a_w — action is all-zeros, so action@a_w==0, unused.
  const float* a_b    = (const float*)d_in[11];
  const float* g1_wl  = (const float*)d_in[12];
  const float* g1_bl  = (const float*)d_in[13];
  const float* g1_wr  = (const float*)d_in[14];
  const float* g1_br  = (const float*)d_in[15];
  const float* g1_we  = (const float*)d_in[16];
  const float* g1_att = (const float*)d_in[17];
  const float* g1_bias= (const float*)d_in[18];
  const float* g2_wl  = (const float*)d_in[19];
  const float* g2_bl  = (const float*)d_in[20];
  const float* g2_wr  = (const float*)d_in[21];
  const float* g2_br  = (const float*)d_in[22];
  const float* g2_we  = (const float*)d_in[23];
  const float* g2_att = (const float*)d_in[24];
  const float* g2_bias= (const float*)d_in[25];

  float* ws = (float*)d_ws;
  // workspace layout (floats)
  float* nodes = ws;                    // 4*197*64   = 50432
  float* U     = nodes + 50432;         // 4*196*800  = 627200
  float* VT    = U     + 627200;        // 4*800*196  = 627200
  float* adjp  = VT    + 627200;        // 4*197*197  = 155236
  float* xl1   = adjp  + 155236;        // 4*197*500  = 394000
  float* xr1   = xl1   + 394000;        // 394000
  float* x1    = xr1   + 394000;        // 394000
  float* xl2   = x1    + 394000;        // 4*197*320  = 252160
  float* xr2   = xl2   + 252160;        // 252160
  float* x2    = xr2   + 252160;        // 252160
  // total ~3.40M floats (~13.6 MB)

  // K1: nodes = [pl ; a_b]
  {
    int tot = 4 * 197 * 64;
    init_nodes_kernel<<<(tot + 255) / 256, 256, 0, stream>>>(latent, a_b, nodes);
  }

  // U = pl @ w1_top + b1   (bias folded here, once)
  wmma_gemm_kernel<<<dim3(25, 13, 4), 32, 0, stream>>>(
      nodes, d0_w1, d0_b1, U, 196, 64, 800, 197 * 64, 196 * 800, 0);
  // VT = (pl @ w1_bot)^T  (transposed store for coalesced pairwise reads)
  wmma_gemm_kernel<<<dim3(25, 13, 4), 32, 0, stream>>>(
      nodes, d0_w1 + 64 * 800, nullptr, VT, 196, 64, 800, 197 * 64, 800 * 196, 1);

  // K3: edge scores + gumbel gate -> padded adjacency
  pairwise_kernel<<<4 * 197, 256, 0, stream>>>(U, VT, d0_w2, d0_b2, gumbel, adjp);

  // GAT layer 1
  wmma_gemm_kernel<<<dim3(16, 13, 4), 32, 0, stream>>>(
      nodes, g1_wl, g1_bl, xl1, 197, 64, 500, 197 * 64, 197 * 500, 0);
  wmma_gemm_kernel<<<dim3(16, 13, 4), 32, 0, stream>>>(
      nodes, g1_wr, g1_br, xr1, 197, 64, 500, 197 * 64, 197 * 500, 0);
  gat_attn_kernel<<<4 * 197, 256, 0, stream>>>(
      xl1, xr1, adjp, g1_att, g1_we, g1_bias, x1, 197, 5, 100, 0.01f);

  // GAT layer 2
  wmma_gemm_kernel<<<dim3(10, 13, 4), 32, 0, stream>>>(
      x1, g2_wl, g2_bl, xl2, 197, 500, 320, 197 * 500, 197 * 320, 0);
  wmma_gemm_kernel<<<dim3(10, 13, 4), 32, 0, stream>>>(
      x1, g2_wr, g2_br, xr2, 197, 500, 320, 197 * 500, 197 * 320, 0);
  gat_attn_kernel<<<4 * 197, 256, 0, stream>>>(
      xl2, xr2, adjp, g2_att, g2_we, g2_bias, x2, 197, 5, 64, 1.0f);

  // final softmax over first 64 channels of first 196 nodes
  final_softmax_kernel<<<4 * 196, 64, 0, stream>>>(x2, (float*)d_out);
}